// SnnCoreDU_4492535792365
// MI455X (gfx1250) — compile-verified
//
#include <hip/hip_runtime.h>

typedef float v2f __attribute__((ext_vector_type(2)));
typedef float v8f __attribute__((ext_vector_type(8)));

#define B_DIM   4096
#define IN_DIM  512
#define H_DIM   1024
#define T_STEPS 50
#define BETA    0.95f

// zero the tot[] region of d_out (harness poisons it with 0xAA)
__global__ void snn_tot_init_kernel(float* __restrict__ tot) {
    int i = blockIdx.x * blockDim.x + threadIdx.x;
    if (i < B_DIM) tot[i] = 0.0f;
}

// Fused: fp32 WMMA GEMM (inp = x@W + b), register-blocked 32x32 per wave
// (2x2 waves -> 64x64 per workgroup for WGP$ reuse), then a register-resident
// 50-step LIF recurrence at 6 VALU instr / 2 elems / step, storing agg and
// reducing tot = 2 * rowsum(agg) (the two LIF layers are identical).
__global__ __launch_bounds__(128)
void snn_fused_kernel(const float* __restrict__ x,
                      const float* __restrict__ W,
                      const float* __restrict__ bias,
                      float* __restrict__ agg,
                      float* __restrict__ tot) {
    const int lane = threadIdx.x & 31;
    const int wave = threadIdx.x >> 5;
    const int half = lane >> 4;                 // 0: lanes 0-15, 1: lanes 16-31
    const int l16  = lane & 15;

    // 64x64 block tile; 4 waves in a 2x2 arrangement of 32x32 sub-tiles
    const int M0 = (blockIdx.x >> 4) * 64;      // 64 M-blocks
    const int N0 = (blockIdx.x & 15) * 64;      // 16 N-blocks
    const int m0 = M0 + (wave >> 1) * 32;
    const int n0 = N0 + (wave & 1) * 32;

    // A 16x4 f32 frag: lanes 0-15 hold K = k,k+1 ; lanes 16-31 hold K = k+2,k+3
    const float* xrow0 = x + (size_t)(m0 + l16) * IN_DIM + 2 * half;
    const float* xrow1 = xrow0 + (size_t)16 * IN_DIM;
    // B 4x16 f32 frag: VGPR0 = row K / K+2, VGPR1 = row K+1 / K+3
    const float* wcol0 = W + (n0 + l16) + (size_t)(2 * half) * H_DIM;
    const float* wcol1 = wcol0 + 16;

    v8f c00 = {0.f,0.f,0.f,0.f,0.f,0.f,0.f,0.f};
    v8f c01 = c00, c10 = c00, c11 = c00;

    #pragma unroll 2
    for (int k = 0; k < IN_DIM; k += 4) {
        v2f a0 = *(const v2f*)(xrow0 + k);       // contiguous 8B loads
        v2f a1 = *(const v2f*)(xrow1 + k);
        v2f b0, b1;
        b0.x = wcol0[(size_t)k * H_DIM];
        b0.y = wcol0[(size_t)k * H_DIM + H_DIM];
        b1.x = wcol1[(size_t)k * H_DIM];
        b1.y = wcol1[(size_t)k * H_DIM + H_DIM];
        c00 = __builtin_amdgcn_wmma_f32_16x16x4_f32(false, a0, false, b0, (short)0, c00, false, false);
        c01 = __builtin_amdgcn_wmma_f32_16x16x4_f32(false, a0, false, b1, (short)0, c01, false, false);
        c10 = __builtin_amdgcn_wmma_f32_16x16x4_f32(false, a1, false, b0, (short)0, c10, false, false);
        c11 = __builtin_amdgcn_wmma_f32_16x16x4_f32(false, a1, false, b1, (short)0, c11, false, false);
    }

    // Per-lane state for 32 elements (4 frags x 8).
    // LIF identities: sel_t = inp - sp_{t-1} in {inp, inp-1};
    // agg = T*inp - sum(sel) + sp_final  (exact integer -> rintf snap).
    const float bb0 = bias[n0 + l16];
    const float bb1 = bias[n0 + 16 + l16];

    float inp[32], inp1[32], mem[32], acc[32], sel[32];
    #pragma unroll
    for (int i = 0; i < 8; ++i) {
        inp[0 * 8 + i] = c00[i] + bb0;
        inp[1 * 8 + i] = c01[i] + bb1;
        inp[2 * 8 + i] = c10[i] + bb0;
        inp[3 * 8 + i] = c11[i] + bb1;
    }
    #pragma unroll
    for (int j = 0; j < 32; ++j) {
        inp1[j] = inp[j] - 1.0f;
        mem[j]  = 0.0f;
        acc[j]  = 0.0f;
        sel[j]  = inp[j];            // sp_{-1} = 0
    }

    for (int t = 0; t < T_STEPS; ++t) {
        #pragma unroll
        for (int j = 0; j < 32; ++j) {
            mem[j] = __builtin_fmaf(BETA, mem[j], sel[j]);   // pk_fma (x2 elems)
            acc[j] += sel[j];                                // pk_add
            sel[j] = (mem[j] > 1.0f) ? inp1[j] : inp[j];     // cmp + cndmask
        }
    }

    // Epilogue: reconstruct agg, store, and reduce row-sums into tot.
    #pragma unroll
    for (int f = 0; f < 4; ++f) {
        const int mi = f >> 1, ni = f & 1;
        const int rowbase = m0 + mi * 16 + half * 8;
        const int col     = n0 + ni * 16 + l16;
        float* aggp = agg + (size_t)rowbase * H_DIM + col;
        #pragma unroll
        for (int i = 0; i < 8; ++i) {
            const int j = f * 8 + i;
            float spl = (mem[j] > 1.0f) ? 1.0f : 0.0f;       // sp at t = T-1
            float ag  = __builtin_rintf(
                __builtin_fmaf((float)T_STEPS, inp[j], -acc[j]) + spl);
            aggp[(size_t)i * H_DIM] = ag;
            float s = ag;
            s += __shfl_xor(s, 1, 32);   // masks < 16: stays within each
            s += __shfl_xor(s, 2, 32);   // 16-lane half (one row per half)
            s += __shfl_xor(s, 4, 32);
            s += __shfl_xor(s, 8, 32);
            if (l16 == 0) {
                atomicAdd(&tot[rowbase + i], 2.0f * s);      // x2: two layers
            }
        }
    }
}

extern "C" void kernel_launch(void* const* d_in, const int* in_sizes, int n_in,
                              void* d_out, int out_size, void* d_ws, size_t ws_size,
                              hipStream_t stream) {
    const float* x    = (const float*)d_in[0];   // [4096, 512]
    const float* W    = (const float*)d_in[1];   // [512, 1024]
    const float* bias = (const float*)d_in[2];   // [1024]
    float* agg = (float*)d_out;                  // [4096, 1024]
    float* tot = agg + (size_t)B_DIM * H_DIM;    // [4096]

    snn_tot_init_kernel<<<B_DIM / 256, 256, 0, stream>>>(tot);

    // (4096/64) * (1024/64) = 1024 blocks, 128 threads (4 waves, 64x64 tile)
    const int nblocks = (B_DIM / 64) * (H_DIM / 64);
    snn_fused_kernel<<<nblocks, 128, 0, stream>>>(x, W, bias, agg, tot);
}